// GCN_9131100472079
// MI455X (gfx1250) — compile-verified
//
#include <hip/hip_runtime.h>

#define HIDDEN 256
#define LIN1 128

typedef __attribute__((ext_vector_type(2))) float v2f;
typedef __attribute__((ext_vector_type(8))) float v8f;
typedef __attribute__((ext_vector_type(4))) int   v4i;

// pointer-to-AS1/AS3 v4i, matching the async-load builtin's prototype
typedef __attribute__((address_space(1))) v4i* as1_v4i_p;
typedef __attribute__((address_space(3))) v4i* as3_v4i_p;

#if __has_builtin(__builtin_amdgcn_global_load_async_to_lds_b128)
#define ASYNC_LDS 1
#else
#define ASYNC_LDS 0
#endif

// ---------------------------------------------------------------------------
// Degree / normalization:  deg = 1 (self loop) + #edges into node; dinv=rsqrt
// ---------------------------------------------------------------------------
__global__ void k_deg_init(float* __restrict__ deg, int n) {
    int i = blockIdx.x * blockDim.x + threadIdx.x;
    if (i < n) deg[i] = 1.0f;  // self loop
}

__global__ void k_deg_accum(const int* __restrict__ dst, float* __restrict__ deg, int e) {
    int i = blockIdx.x * blockDim.x + threadIdx.x;
    if (i < e) atomicAdd(&deg[dst[i]], 1.0f);
}

__global__ void k_rsqrt_inplace(float* __restrict__ d, int n) {
    int i = blockIdx.x * blockDim.x + threadIdx.x;
    if (i < n) d[i] = __frsqrt_rn(d[i]);  // deg >= 1 always (self loops)
}

__global__ void k_edge_norm(const int* __restrict__ src, const int* __restrict__ dst,
                            const float* __restrict__ dinv, float* __restrict__ norm, int e) {
    int i = blockIdx.x * blockDim.x + threadIdx.x;
    if (i < e) norm[i] = dinv[src[i]] * dinv[dst[i]];
}

// ---------------------------------------------------------------------------
// W transpose (256x256) so WMMA B-fragments become single aligned b64 loads.
// ---------------------------------------------------------------------------
__global__ void k_transpose(const float* __restrict__ W, float* __restrict__ Wt) {
    int i = blockIdx.x * blockDim.x + threadIdx.x;   // HIDDEN*HIDDEN threads
    int r = i >> 8, c = i & (HIDDEN - 1);
    Wt[(size_t)c * HIDDEN + r] = W[i];
}

// ---------------------------------------------------------------------------
// Layer 1 GEMM:  out[n, 0:256] = x[n,0]*W1[0,:] + x[n,1]*W1[1,:]   (K=2)
// ---------------------------------------------------------------------------
__global__ void k_gemm1(const float* __restrict__ x, const float* __restrict__ W1,
                        float* __restrict__ out, int n) {
    int gid = blockIdx.x * blockDim.x + threadIdx.x;
    int node = gid >> 6;
    int c4 = (gid & 63) << 2;
    if (node >= n) return;
    float x0 = x[(size_t)node * 2 + 0];
    float x1 = x[(size_t)node * 2 + 1];
    float4 w0 = *(const float4*)(W1 + c4);
    float4 w1 = *(const float4*)(W1 + HIDDEN + c4);
    float4 r;
    r.x = fmaf(x0, w0.x, x1 * w1.x);
    r.y = fmaf(x0, w0.y, x1 * w1.y);
    r.z = fmaf(x0, w0.z, x1 * w1.z);
    r.w = fmaf(x0, w0.w, x1 * w1.w);
    *(float4*)(out + (size_t)node * HIDDEN + c4) = r;
}

// ---------------------------------------------------------------------------
// WMMA fp32 GEMM:  C[n,256] = relu(A[n,256]) @ W  (W passed pre-transposed)
// 512 threads = 16 waves per block; block covers one 16-row strip; wave w owns
// a 16-col tile. A tile async-copied to LDS (ASYNCcnt path) when available,
// stride-260 padding keeps the v_wmma A-fragment ds reads bank-conflict-free.
// ReLU is fused into the fragment read (co-executes with the WMMA pipe).
// ---------------------------------------------------------------------------
__global__ __launch_bounds__(512) void k_gemm_wmma(
    const float* __restrict__ A, const float* __restrict__ Wt,
    float* __restrict__ C, int n)
{
    __shared__ float As[16][HIDDEN + 4];   // +4 pad -> 32 distinct banks on frag reads

    const int row0 = blockIdx.x << 4;
    const int tid = threadIdx.x;
    const bool full = (row0 + 16 <= n);

#if ASYNC_LDS
    if (full) {
        // 1024 x 16B chunks, 2 per thread, via the async Global->LDS engine
        for (int i = tid; i < 16 * (HIDDEN / 4); i += 512) {
            int r  = i >> 6;
            int c4 = (i & 63) << 2;
            const float* gp = A + (size_t)(row0 + r) * HIDDEN + c4;
            __builtin_amdgcn_global_load_async_to_lds_b128(
                (as1_v4i_p)(uintptr_t)gp,
                (as3_v4i_p)(uint32_t)(uintptr_t)&As[r][c4],
                0, 0);
        }
#if __has_builtin(__builtin_amdgcn_s_wait_asynccnt)
        __builtin_amdgcn_s_wait_asynccnt(0);
#else
        asm volatile("s_wait_asynccnt 0x0" ::: "memory");
#endif
    } else
#endif
    {
        for (int i = tid; i < 16 * (HIDDEN / 4); i += 512) {
            int r  = i >> 6;
            int c4 = (i & 63) << 2;
            float4 v = {0.f, 0.f, 0.f, 0.f};
            if (row0 + r < n)
                v = *(const float4*)(A + (size_t)(row0 + r) * HIDDEN + c4);
            As[r][c4 + 0] = v.x; As[r][c4 + 1] = v.y;
            As[r][c4 + 2] = v.z; As[r][c4 + 3] = v.w;
        }
    }
    __syncthreads();

    const int wave = tid >> 5;          // 0..15 -> 16-col tile
    const int lane = tid & 31;
    const int col0 = wave << 4;
    const int m    = lane & 15;         // row of A frag / col of B frag
    const int kh   = (lane >> 4) << 1;  // lanes 16-31 hold K+2, K+3
    const float* wrow = Wt + (size_t)(col0 + m) * HIDDEN;  // Wt[col][k]

    v8f acc = {};
    #pragma unroll 8
    for (int k = 0; k < HIDDEN; k += 4) {
        // A frag (16x4 fp32 layout): lane holds relu(A[m][k+kh]), relu(A[m][k+kh+1])
        v2f a = *(const v2f*)(&As[m][k + kh]);
        a.x = fmaxf(a.x, 0.f);
        a.y = fmaxf(a.y, 0.f);
        // B frag: lane holds W[k+kh][col0+m], W[k+kh+1][col0+m] = Wt[col][k+kh..+1]
        v2f b = *(const v2f*)(wrow + k + kh);
        acc = __builtin_amdgcn_wmma_f32_16x16x4_f32(
                  false, a, false, b, (short)0, acc, false, false);
    }

    // C/D layout: vgpr v, lanes 0-15 -> M=v; lanes 16-31 -> M=v+8; N=lane&15
    const int mbase = (lane >> 4) << 3;
    if (full) {
        float* cp = C + (size_t)(row0 + mbase) * HIDDEN + col0 + m;
        #pragma unroll
        for (int v = 0; v < 8; ++v)
            cp[(size_t)v * HIDDEN] = acc[v];
    } else {
        #pragma unroll
        for (int v = 0; v < 8; ++v) {
            int row = row0 + mbase + v;
            if (row < n)
                C[(size_t)row * HIDDEN + col0 + m] = acc[v];
        }
    }
}

// ---------------------------------------------------------------------------
// agg init: self-loop contribution + bias:  agg[n,c] = b[c] + hw[n,c]*dinv[n]^2
// ---------------------------------------------------------------------------
__global__ void k_self_bias(const float* __restrict__ hw, const float* __restrict__ dinv,
                            const float* __restrict__ bias, float* __restrict__ agg, int n) {
    int gid = blockIdx.x * blockDim.x + threadIdx.x;
    int node = gid >> 6;
    int c4 = (gid & 63) << 2;
    if (node >= n) return;
    float di = dinv[node];
    float s = di * di;
    float4 h = *(const float4*)(hw + (size_t)node * HIDDEN + c4);
    float4 b = *(const float4*)(bias + c4);
    float4 r;
    r.x = fmaf(h.x, s, b.x); r.y = fmaf(h.y, s, b.y);
    r.z = fmaf(h.z, s, b.z); r.w = fmaf(h.w, s, b.w);
    *(float4*)(agg + (size_t)node * HIDDEN + c4) = r;
}

// ---------------------------------------------------------------------------
// Edge scatter:  agg[dst] += hw[src] * norm[e]  — one wave32 per edge,
// each lane owns 8 columns (2x float4 loads, 8 f32 atomics). Memory-bound.
// ---------------------------------------------------------------------------
__global__ __launch_bounds__(256) void k_scatter(
    const float* __restrict__ hw, const int* __restrict__ src, const int* __restrict__ dst,
    const float* __restrict__ norm, float* __restrict__ agg, int e)
{
    int gid = blockIdx.x * blockDim.x + threadIdx.x;
    int edge = gid >> 5;
    int lane = gid & 31;
    if (edge >= e) return;
    int s = src[edge];
    int d = dst[edge];
    float w = norm[edge];
    const float4* hp = (const float4*)(hw + (size_t)s * HIDDEN);
    float* ap = agg + (size_t)d * HIDDEN + (lane << 3);
    float4 v0 = hp[lane * 2 + 0];
    float4 v1 = hp[lane * 2 + 1];
    atomicAdd(ap + 0, v0.x * w); atomicAdd(ap + 1, v0.y * w);
    atomicAdd(ap + 2, v0.z * w); atomicAdd(ap + 3, v0.w * w);
    atomicAdd(ap + 4, v1.x * w); atomicAdd(ap + 5, v1.y * w);
    atomicAdd(ap + 6, v1.z * w); atomicAdd(ap + 7, v1.w * w);
}

// ---------------------------------------------------------------------------
// Global mean pool
// ---------------------------------------------------------------------------
__global__ void k_pool_zero(float* __restrict__ sums, float* __restrict__ cnt, int g) {
    int i = blockIdx.x * blockDim.x + threadIdx.x;
    if (i < g * HIDDEN) sums[i] = 0.f;
    if (i < g) cnt[i] = 0.f;
}

__global__ __launch_bounds__(256) void k_pool(
    const float* __restrict__ h, const int* __restrict__ batch,
    float* __restrict__ sums, float* __restrict__ cnt, int n)
{
    int gid = blockIdx.x * blockDim.x + threadIdx.x;
    int node = gid >> 5;
    int lane = gid & 31;
    if (node >= n) return;
    int g = batch[node];
    const float4* hp = (const float4*)(h + (size_t)node * HIDDEN);
    float* sp = sums + (size_t)g * HIDDEN + (lane << 3);
    float4 v0 = hp[lane * 2 + 0];
    float4 v1 = hp[lane * 2 + 1];
    atomicAdd(sp + 0, v0.x); atomicAdd(sp + 1, v0.y);
    atomicAdd(sp + 2, v0.z); atomicAdd(sp + 3, v0.w);
    atomicAdd(sp + 4, v1.x); atomicAdd(sp + 5, v1.y);
    atomicAdd(sp + 6, v1.z); atomicAdd(sp + 7, v1.w);
    if (lane == 0) atomicAdd(&cnt[g], 1.f);
}

// ---------------------------------------------------------------------------
// Head MLP: g = relu(mean @ lin1 + b1); out = g @ lin2 + b2
// ---------------------------------------------------------------------------
__global__ __launch_bounds__(LIN1) void k_mlp(
    const float* __restrict__ sums, const float* __restrict__ cnt,
    const float* __restrict__ w1, const float* __restrict__ b1,
    const float* __restrict__ w2, const float* __restrict__ b2,
    float* __restrict__ out)
{
    __shared__ float red[LIN1];
    int g = blockIdx.x;
    int j = threadIdx.x;
    float inv = 1.0f / fmaxf(cnt[g], 1.0f);
    const float* sg = sums + (size_t)g * HIDDEN;
    float acc = b1[j];
    #pragma unroll 8
    for (int c = 0; c < HIDDEN; ++c)
        acc = fmaf(sg[c] * inv, w1[(size_t)c * LIN1 + j], acc);
    acc = fmaxf(acc, 0.0f);
    red[j] = acc * w2[j];
    __syncthreads();
    for (int s = LIN1 / 2; s > 0; s >>= 1) {
        if (j < s) red[j] += red[j + s];
        __syncthreads();
    }
    if (j == 0) out[g] = red[0] + b2[0];
}

// ---------------------------------------------------------------------------
extern "C" void kernel_launch(void* const* d_in, const int* in_sizes, int n_in,
                              void* d_out, int out_size, void* d_ws, size_t ws_size,
                              hipStream_t stream) {
    const float* x      = (const float*)d_in[0];
    const int*   ei     = (const int*)d_in[1];
    const int*   batch  = (const int*)d_in[2];
    const float* W[4]   = {(const float*)d_in[3], (const float*)d_in[5],
                           (const float*)d_in[7], (const float*)d_in[9]};
    const float* b[4]   = {(const float*)d_in[4], (const float*)d_in[6],
                           (const float*)d_in[8], (const float*)d_in[10]};
    const float* lin1_w = (const float*)d_in[11];
    const float* lin1_b = (const float*)d_in[12];
    const float* lin2_w = (const float*)d_in[13];
    const float* lin2_b = (const float*)d_in[14];
    float* out = (float*)d_out;

    const int n = in_sizes[0] / 2;       // nodes
    const int e = in_sizes[1] / 2;       // edges
    const int g = out_size;              // graphs
    const int* src = ei;
    const int* dst = ei + e;

    // workspace carve-out
    char* w8 = (char*)d_ws;
    size_t off = 0;
    auto carve = [&](size_t bytes) {
        void* p = w8 + off;
        off += (bytes + 255) & ~(size_t)255;
        return p;
    };
    const size_t nh = (size_t)n * HIDDEN;
    float* dinv = (float*)carve((size_t)n * 4);
    float* norm = (float*)carve((size_t)e * 4);
    float* buf0 = (float*)carve(nh * 4);                    // agg / next input
    float* buf1 = (float*)carve(nh * 4);                    // GEMM output h@W
    float* Wt   = (float*)carve((size_t)HIDDEN * HIDDEN * 4);  // transposed W
    float* sums = (float*)carve((size_t)g * HIDDEN * 4);
    float* cnt  = (float*)carve((size_t)g * 4);
    (void)ws_size; (void)n_in;

    const int B = 256;
    // 1) degrees & symmetric normalization
    k_deg_init<<<(n + B - 1) / B, B, 0, stream>>>(dinv, n);
    k_deg_accum<<<(e + B - 1) / B, B, 0, stream>>>(dst, dinv, e);
    k_rsqrt_inplace<<<(n + B - 1) / B, B, 0, stream>>>(dinv, n);
    k_edge_norm<<<(e + B - 1) / B, B, 0, stream>>>(src, dst, dinv, norm, e);

    const int gN64   = (int)(((size_t)n * 64 + B - 1) / B);
    const int gE32   = (int)(((size_t)e * 32 + B - 1) / B);
    const int gTiles = (n + 15) / 16;
    const int gTr    = (HIDDEN * HIDDEN) / B;

    // 2) layer 1 (K=2 GEMM) + aggregate
    k_gemm1<<<gN64, B, 0, stream>>>(x, W[0], buf1, n);
    k_self_bias<<<gN64, B, 0, stream>>>(buf1, dinv, b[0], buf0, n);
    k_scatter<<<gE32, B, 0, stream>>>(buf1, src, dst, norm, buf0, e);

    // 3) layers 2..4: fused ReLU + fp32 WMMA GEMM (W pre-transposed), aggregate
    for (int l = 1; l < 4; ++l) {
        k_transpose<<<gTr, B, 0, stream>>>(W[l], Wt);
        k_gemm_wmma<<<gTiles, 512, 0, stream>>>(buf0, Wt, buf1, n);
        k_self_bias<<<gN64, B, 0, stream>>>(buf1, dinv, b[l], buf0, n);
        k_scatter<<<gE32, B, 0, stream>>>(buf1, src, dst, norm, buf0, e);
    }

    // 4) mean pool + head MLP
    k_pool_zero<<<(g * HIDDEN + B - 1) / B, B, 0, stream>>>(sums, cnt, g);
    k_pool<<<(int)(((size_t)n * 32 + B - 1) / B), B, 0, stream>>>(buf0, batch, sums, cnt, n);
    k_mlp<<<g, LIN1, 0, stream>>>(sums, cnt, lin1_w, lin1_b, lin2_w, lin2_b, out);
}